// SparseReservoir_54855322305129
// MI455X (gfx1250) — compile-verified
//
#include <hip/hip_runtime.h>
#include <hip/hip_bf16.h>
#include <math.h>

#define N_RES 8192
#define N_IN  128
#define BATCH 16

typedef __attribute__((ext_vector_type(2))) float v2f;
typedef __attribute__((ext_vector_type(8))) float v8f;

// ws layout (floats):
//   zT     [N_RES][16]   @ 0        (131072)  accumulator, transposed
//   stateT [N_RES][16]   @ 131072   (131072)
//   xT     [N_IN ][16]   @ 262144   (2048)
//   Wd     [N_RES][N_IN] @ 264192   (1048576) dense W_in (only if ws big enough)
#define ZT_OFF     0
#define STATET_OFF 131072
#define XT_OFF     262144
#define WD_OFF     264192
#define WS_FLOATS_FULL (WD_OFF + N_RES * N_IN)
#define WS_FLOATS_MIN  WD_OFF

// ---------------------------------------------------------------- init
__global__ void res_init_kernel(const float* __restrict__ state,
                                const float* __restrict__ x,
                                const float* __restrict__ res_bias,
                                const float* __restrict__ in_bias,
                                float* __restrict__ ws, int zero_wd) {
  int idx = blockIdx.x * blockDim.x + threadIdx.x;
  if (idx < N_RES * BATCH) {
    int c = idx >> 4, b = idx & 15;
    ws[ZT_OFF + idx]     = res_bias[c] + in_bias[c];
    ws[STATET_OFF + idx] = state[b * N_RES + c];
  }
  if (idx < N_IN * BATCH) {
    int c = idx >> 4, b = idx & 15;
    ws[XT_OFF + idx] = x[b * N_IN + c];
  }
  if (zero_wd && idx < N_RES * N_IN) {
    ws[WD_OFF + idx] = 0.0f;
  }
}

// ------------------------------------------- scatter W_in into dense 4MB
__global__ void res_scatter_win_kernel(const float* __restrict__ vals,
                                       const int* __restrict__ rows,
                                       const int* __restrict__ cols,
                                       float* __restrict__ Wd, int nnz) {
  int i = blockIdx.x * blockDim.x + threadIdx.x;
  if (i < nnz) {
    unsafeAtomicAdd(&Wd[rows[i] * N_IN + cols[i]], vals[i]);
  }
}

// --------------------------------- z_in += Wd(16 rows) x x^T via WMMA f32
// one wave per 16-row tile; 512 tiles total; EXEC all-ones (no divergence)
__global__ void res_zin_wmma_kernel(const float* __restrict__ Wd,
                                    const float* __restrict__ xT,
                                    float* __restrict__ zT) {
  int wave = (blockIdx.x * blockDim.x + threadIdx.x) >> 5; // tile id 0..511
  int lane = threadIdx.x & 31;
  int lm = lane & 15;
  int hi = lane >> 4;
  int r0 = wave << 4;

#if __has_builtin(__builtin_amdgcn_wmma_f32_16x16x4_f32)
  v8f c = {};
  for (int k0 = 0; k0 < N_IN; k0 += 4) {
    v2f a, b;
    // A 16x4: lanes 0-15 rows M, VGPR0/1 = K0,K1; lanes 16-31 = K2,K3
    a.x = Wd[(r0 + lm) * N_IN + k0 + 2 * hi + 0];
    a.y = Wd[(r0 + lm) * N_IN + k0 + 2 * hi + 1];
    // B 4x16: lanes 0-15 cols N, VGPR0/1 = K0,K1; lanes 16-31 = K2,K3
    b.x = xT[(k0 + 2 * hi + 0) * 16 + lm];
    b.y = xT[(k0 + 2 * hi + 1) * 16 + lm];
    c = __builtin_amdgcn_wmma_f32_16x16x4_f32(
        /*neg_a=*/false, a, /*neg_b=*/false, b,
        /*c_mod=*/(short)0, c, /*reuse_a=*/false, /*reuse_b=*/false);
  }
  // C/D 16x16: VGPR j -> row r0+j (lanes 0-15) / r0+j+8 (lanes 16-31)
  for (int j = 0; j < 8; ++j) {
    int r = r0 + j + 8 * hi;
    zT[r * 16 + lm] += c[j];
  }
#else
  for (int j = 0; j < 8; ++j) {
    int r = r0 + j + 8 * hi;
    float acc = 0.0f;
    for (int k = 0; k < N_IN; ++k) acc += Wd[r * N_IN + k] * xT[k * 16 + lm];
    zT[r * 16 + lm] += acc;
  }
#endif
}

// ------------------------- fallback z_in path (atomics) if ws too small
__global__ void res_scatter_zin_kernel(const float* __restrict__ vals,
                                       const int* __restrict__ rows,
                                       const int* __restrict__ cols,
                                       const float* __restrict__ xT,
                                       float* __restrict__ zT, int nnz) {
  int t = blockIdx.x * blockDim.x + threadIdx.x;
  int i = t >> 4;
  int b = t & 15;
  if (i < nnz) {
    unsafeAtomicAdd(&zT[rows[i] * 16 + b], vals[i] * xT[cols[i] * 16 + b]);
  }
}

// ------------------------------------ main reservoir scatter (6.7M nnz)
__global__ void res_scatter_res_kernel(const float* __restrict__ vals,
                                       const int* __restrict__ rows,
                                       const int* __restrict__ cols,
                                       const float* __restrict__ stateT,
                                       float* __restrict__ zT, int nnz) {
  int t = blockIdx.x * blockDim.x + threadIdx.x;
  int i = t >> 4;   // nnz index (16 lanes per nnz, one per batch)
  int b = t & 15;   // batch lane
  if (i < nnz) {
    if (b == 0 && i + 4096 < nnz) {
      __builtin_prefetch(&vals[i + 4096], 0, 0);  // global_prefetch_b8
      __builtin_prefetch(&rows[i + 4096], 0, 0);
      __builtin_prefetch(&cols[i + 4096], 0, 0);
    }
    int r = rows[i];
    int c = cols[i];
    float v = vals[i];
    // coalesced 64B gather across the 16 lanes; coalesced L2 atomics
    unsafeAtomicAdd(&zT[r * 16 + b], v * stateT[c * 16 + b]);
  }
}

// ----------------------------------------------------------- finalize
__global__ void res_finalize_kernel(const float* __restrict__ state,
                                    const float* __restrict__ zT,
                                    float* __restrict__ out) {
  int idx = blockIdx.x * blockDim.x + threadIdx.x;
  if (idx < BATCH * N_RES) {
    int b = idx >> 13;            // / N_RES
    int r = idx & (N_RES - 1);
    out[idx] = (1.0f - 0.9f) * state[idx] + 0.9f * erff(zT[r * 16 + b]);
  }
}

extern "C" void kernel_launch(void* const* d_in, const int* in_sizes, int n_in,
                              void* d_out, int out_size, void* d_ws, size_t ws_size,
                              hipStream_t stream) {
  const float* state    = (const float*)d_in[0];
  const float* x        = (const float*)d_in[1];
  const float* res_vals = (const float*)d_in[2];
  const int*   res_rows = (const int*)d_in[3];
  const int*   res_cols = (const int*)d_in[4];
  const float* res_bias = (const float*)d_in[5];
  const float* in_vals  = (const float*)d_in[6];
  const int*   in_rows  = (const int*)d_in[7];
  const int*   in_cols  = (const int*)d_in[8];
  const float* in_bias  = (const float*)d_in[9];
  float* out = (float*)d_out;

  const int res_nnz = in_sizes[2];
  const int in_nnz  = in_sizes[6];

  float* ws = (float*)d_ws;
  float* zT     = ws + ZT_OFF;
  float* stateT = ws + STATET_OFF;
  float* xT     = ws + XT_OFF;
  float* Wd     = ws + WD_OFF;

  const bool have_wd = ws_size >= (size_t)WS_FLOATS_FULL * sizeof(float);

  // 1) init accumulator (= biases), transpose state and x; zero dense W_in
  {
    int n = have_wd ? (N_RES * N_IN) : (N_RES * BATCH);
    res_init_kernel<<<(n + 255) / 256, 256, 0, stream>>>(
        state, x, res_bias, in_bias, ws, have_wd ? 1 : 0);
  }

  // 2) z_in term
  if (have_wd) {
    res_scatter_win_kernel<<<(in_nnz + 255) / 256, 256, 0, stream>>>(
        in_vals, in_rows, in_cols, Wd, in_nnz);
    // 512 tiles, 8 waves (tiles) per 256-thread block -> 64 blocks
    res_zin_wmma_kernel<<<(N_RES / 16) / 8, 256, 0, stream>>>(Wd, xT, zT);
  } else {
    long long t = (long long)in_nnz * 16;
    res_scatter_zin_kernel<<<(unsigned)((t + 255) / 256), 256, 0, stream>>>(
        in_vals, in_rows, in_cols, xT, zT, in_nnz);
  }

  // 3) reservoir term: 16 lanes (batch) per nnz
  {
    long long t = (long long)res_nnz * 16;
    res_scatter_res_kernel<<<(unsigned)((t + 255) / 256), 256, 0, stream>>>(
        res_vals, res_rows, res_cols, stateT, zT, res_nnz);
  }

  // 4) leaky erf
  res_finalize_kernel<<<(BATCH * N_RES + 255) / 256, 256, 0, stream>>>(
      state, zT, out);
}